// RadarElevationLearner_12300786336439
// MI455X (gfx1250) — compile-verified
//
#include <hip/hip_runtime.h>
#include <stdint.h>

// ---------------------------------------------------------------------------
// RadarElevationLearner for MI455X (gfx1250).
// Streaming gumbel-argmax kernel: 207MB of gumbel_u is the dominant cost
// (~9us at 23.3 TB/s), plus ~200M transcendentals. No matrix products exist
// in this workload (rank-1 attention scores; one-hot einsum == scatter), so
// the CDNA5 feature used is the async global->LDS DMA path:
// global_load_async_to_lds_b128 + s_wait_asynccnt double-buffering each
// 3600B gumbel row per wave, overlapping HBM streaming with log/exp compute.
// ---------------------------------------------------------------------------

#define S_LEN 900
#define ITERS 29          // ceil(900/32) elements per lane

__device__ __forceinline__ void async_row_prefetch(const float* __restrict__ grow,
                                                   uint32_t ldsBase, int lane) {
  // 3600 bytes = 225 x b128 lane-transfers: 7 full wave issues + 1 lane-0 tail.
  const char* g = (const char*)grow;
#pragma unroll
  for (int it = 0; it < 7; ++it) {
    int slot = it * 32 + lane;
    uint64_t ga = (uint64_t)(uintptr_t)(g + slot * 16);
    uint32_t la = ldsBase + (uint32_t)(slot * 16);
    asm volatile("global_load_async_to_lds_b128 %0, %1, off"
                 :: "v"(la), "v"(ga) : "memory");
  }
  if (lane == 0) {
    uint64_t ga = (uint64_t)(uintptr_t)(g + 224 * 16);
    uint32_t la = ldsBase + (uint32_t)(224 * 16);
    asm volatile("global_load_async_to_lds_b128 %0, %1, off"
                 :: "v"(la), "v"(ga) : "memory");
  }
}

__device__ __forceinline__ void wait_async_le8() {
  asm volatile("s_wait_asynccnt 0x8" ::: "memory");
}
__device__ __forceinline__ void wait_async_0() {
  asm volatile("s_wait_asynccnt 0x0" ::: "memory");
}

// grid: 64 n * 16 blocks, block: 256 threads (8 wave32)
__global__ __launch_bounds__(256) void radar_gumbel_main(
    const float* __restrict__ radar, const float* __restrict__ mde,
    const float* __restrict__ gum,
    const float* __restrict__ pWq, const float* __restrict__ pbq,
    const float* __restrict__ pWk, const float* __restrict__ pbk,
    int* __restrict__ outIdx, float* __restrict__ outC) {
  const int tid  = threadIdx.x;
  const int lane = tid & 31;
  const int wave = tid >> 5;
  const int n      = blockIdx.x >> 4;   // 16 blocks per n
  const int blkInN = blockIdx.x & 15;
  const int b = n >> 3, w = n & 7;
  const long rbase = (long)(w * 8 + b) * S_LEN;   // radar/mde layout [W,B,30,30]

  const float wq = *pWq, bq = *pbq, wk = *pWk, bk = *pbk;

  __shared__ __align__(16) float gbuf[8][2][S_LEN];  // per-wave double buffer
  __shared__ __align__(16) float kArr[S_LEN];
  __shared__ float wmaxS[8], wminS[8];

  const int waveInN = blkInN * 8 + wave;                  // 0..127
  const int nrows   = (S_LEN - waveInN + 127) >> 7;       // 7 or 8 rows/wave
  const float* gumN = gum + (long)n * S_LEN * S_LEN;

  uint32_t ldsBase0 = (uint32_t)(uintptr_t)&gbuf[wave][0][0];
  uint32_t ldsBase1 = (uint32_t)(uintptr_t)&gbuf[wave][1][0];

  // Kick off first row's DMA before k staging: overlaps with the setup phase.
  async_row_prefetch(gumN + (long)waveInN * S_LEN, ldsBase0, lane);

  // --- stage k row in LDS, block-reduce kmax/kmin (for rank-1 softmax max) ---
  float lmax = -__builtin_inff(), lmin = __builtin_inff();
  for (int j = tid; j < S_LEN; j += 256) {
    float kv = mde[rbase + j] * wk + bk;
    kArr[j] = kv;
    lmax = fmaxf(lmax, kv);
    lmin = fminf(lmin, kv);
  }
#pragma unroll
  for (int off = 16; off >= 1; off >>= 1) {
    lmax = fmaxf(lmax, __shfl_xor(lmax, off, 32));
    lmin = fminf(lmin, __shfl_xor(lmin, off, 32));
  }
  if (lane == 0) { wmaxS[wave] = lmax; wminS[wave] = lmin; }
  __syncthreads();
  float kmax = wmaxS[0], kmin = wminS[0];
#pragma unroll
  for (int i = 1; i < 8; ++i) {
    kmax = fmaxf(kmax, wmaxS[i]);
    kmin = fminf(kmin, wminS[i]);
  }

  // Per-lane j-set is identical for every row of this wave: cache k in regs.
  float kreg[ITERS];
#pragma unroll
  for (int ii = 0; ii < ITERS; ++ii) {
    int j = ii * 32 + lane;
    kreg[ii] = (j < S_LEN) ? kArr[j] : 0.0f;
  }

  int buf = 0;
  for (int i = 0; i < nrows; ++i) {
    const int t = waveInN + (i << 7);
    if (i + 1 < nrows) {
      async_row_prefetch(gumN + (long)(t + 128) * S_LEN,
                         (buf ? ldsBase0 : ldsBase1), lane);
      wait_async_le8();   // current buffer's 8 DMAs done; next row in flight
    } else {
      wait_async_0();
    }

    const float r_t = radar[rbase + t];
    const float q   = r_t * wq + bq;
    const float m   = (q >= 0.0f) ? q * kmax : q * kmin;  // exact row max of q*k

    // pass A: attn numerators + row sum
    float e[ITERS];
    float lsum = 0.0f;
#pragma unroll
    for (int ii = 0; ii < ITERS; ++ii) {
      int j = ii * 32 + lane;
      float ev = 0.0f;
      if (j < S_LEN) ev = __expf(q * kreg[ii] - m);
      e[ii] = ev;
      lsum += ev;
    }
#pragma unroll
    for (int off = 16; off >= 1; off >>= 1) lsum += __shfl_xor(lsum, off, 32);
    const float invL = 1.0f / lsum;

    // pass B: z = attn + gumbel noise; running (max, first-index)
    float zmax = -__builtin_inff();
    int jbest = S_LEN;
#pragma unroll
    for (int ii = 0; ii < ITERS; ++ii) {
      int j = ii * 32 + lane;
      float z;
      if (j < S_LEN) {
        float u  = gbuf[wave][buf][j];
        float nz = -__logf(-__logf(u + 1e-8f) + 1e-8f);
        z = e[ii] * invL + nz;
        if (z > zmax) { zmax = z; jbest = j; }
      } else {
        z = -__builtin_inff();
      }
      e[ii] = z;
    }
#pragma unroll
    for (int off = 16; off >= 1; off >>= 1) {
      float oz = __shfl_xor(zmax, off, 32);
      int   oj = __shfl_xor(jbest, off, 32);
      if (oz > zmax || (oz == zmax && oj < jbest)) { zmax = oz; jbest = oj; }
    }

    // pass C: softmax denominator at max; straight-through value
    float ssum = 0.0f;
#pragma unroll
    for (int ii = 0; ii < ITERS; ++ii) ssum += __expf(e[ii] - zmax);
#pragma unroll
    for (int off = 16; off >= 1; off >>= 1) ssum += __shfl_xor(ssum, off, 32);
    const float p = 1.0f / ssum;
    const float y = (1.0f - p) + p;
    const float src = (r_t != 0.0f) ? r_t : 0.0f;

    if (lane == 0) {
      outIdx[n * S_LEN + t] = jbest;
      outC[n * S_LEN + t]   = y * src;
    }
    buf ^= 1;
  }
}

// grid: 64 (one per n), block: 256. Deterministic gather (no atomics) +
// residual + transpose to [B,1,S,W].
__global__ __launch_bounds__(256) void radar_gumbel_final(
    const float* __restrict__ radar, const int* __restrict__ outIdx,
    const float* __restrict__ outC, const float* __restrict__ pScale,
    const float* __restrict__ pBeta, float* __restrict__ out) {
  const int n = blockIdx.x;
  const int b = n >> 3, w = n & 7;
  const long rbase = (long)(w * 8 + b) * S_LEN;

  __shared__ int   sIdx[S_LEN];
  __shared__ float sC[S_LEN];
  for (int t = threadIdx.x; t < S_LEN; t += 256) {
    sIdx[t] = outIdx[n * S_LEN + t];
    sC[t]   = outC[n * S_LEN + t];
  }
  __syncthreads();

  // attended_out == ln_beta exactly (LayerNorm over size-1 axis)
  const float add = (*pScale) * (*pBeta);

  const int s0 = threadIdx.x;
  const int s1 = s0 + 256, s2 = s0 + 512, s3 = s0 + 768;
  float a0 = 0.0f, a1 = 0.0f, a2 = 0.0f, a3 = 0.0f;
  for (int t = 0; t < S_LEN; ++t) {     // ascending t: deterministic sum order
    int   ix = sIdx[t];
    float cv = sC[t];
    if (ix == s0) a0 += cv;
    if (ix == s1) a1 += cv;
    if (ix == s2) a2 += cv;
    if (ix == s3) a3 += cv;
  }

  int   ss[4] = {s0, s1, s2, s3};
  float aa[4] = {a0, a1, a2, a3};
#pragma unroll
  for (int k = 0; k < 4; ++k) {
    int s = ss[k];
    if (s < S_LEN) {
      float o = aa[k];
      float r = radar[rbase + s];
      float fin = o;
      if (r != 0.0f && o != 0.0f) fin = o + add;   // combined * scale * ln_beta
      out[((long)b * S_LEN + s) * 8 + w] = fin;
    }
  }
}

extern "C" void kernel_launch(void* const* d_in, const int* in_sizes, int n_in,
                              void* d_out, int out_size, void* d_ws, size_t ws_size,
                              hipStream_t stream) {
  (void)in_sizes; (void)n_in; (void)out_size; (void)ws_size;
  const float* radar = (const float*)d_in[0];
  const float* mde   = (const float*)d_in[1];
  const float* gum   = (const float*)d_in[2];
  const float* Wq    = (const float*)d_in[3];
  const float* bq    = (const float*)d_in[4];
  const float* Wk    = (const float*)d_in[5];
  const float* bk    = (const float*)d_in[6];
  // d_in[7..10] = Wv,bv,Wo,bo: dead (LayerNorm over size-1 axis kills them)
  const float* lnBeta = (const float*)d_in[12];
  const float* scale  = (const float*)d_in[13];
  float* out = (float*)d_out;

  int*   wsIdx = (int*)d_ws;                                       // 57600 ints
  float* wsC   = (float*)((char*)d_ws + 64 * S_LEN * sizeof(int)); // 57600 floats

  radar_gumbel_main<<<dim3(64 * 16), dim3(256), 0, stream>>>(
      radar, mde, gum, Wq, bq, Wk, bk, wsIdx, wsC);
  radar_gumbel_final<<<dim3(64), dim3(256), 0, stream>>>(
      radar, wsIdx, wsC, scale, lnBeta, out);
}